// MyConv_11373073399926
// MI455X (gfx1250) — compile-verified
//
#include <hip/hip_runtime.h>
#include <hip/hip_bf16.h>

// ---------------------------------------------------------------------------
// Problem geometry (compile-time constants from the reference)
// ---------------------------------------------------------------------------
#define N_BATCH 8
#define C_IN    64
#define F_OUT   64
#define H_IMG   32
#define W_IMG   32
#define KDIM    (C_IN * 9)                   // 576 = GEMM K (reordered: k = tap*64 + c)
#define EPS_F   1e-8f
#define OUT_ELEMS (N_BATCH * F_OUT * H_IMG * W_IMG) // 524288
#define NBINS   9

typedef float v2f __attribute__((ext_vector_type(2)));
typedef float v8f __attribute__((ext_vector_type(8)));

// gfx1250 async global->LDS fill (ASYNCcnt path, CDNA5 ISA)
__device__ __forceinline__ void async_f32_to_lds(unsigned lds_byte_off,
                                                 const float* gptr)
{
    asm volatile("global_load_async_to_lds_b32 %0, %1, off"
                 :
                 : "v"(lds_byte_off),
                   "v"((unsigned long long)(uintptr_t)gptr)
                 : "memory");
}

__device__ __forceinline__ void wait_asynccnt0()
{
    asm volatile("s_wait_asynccnt 0" ::: "memory");
}

// ---------------------------------------------------------------------------
// Fused kernel: one 256-thread block per (n,y) image row.
//  LDS layout (221 KB, CDNA5 WGP allows up to 320 KB/workgroup):
//    wt[f][k]   : w repacked K-major (k = tap*64 + c)        -> A pairs contiguous
//    xb[xcol][k]: padded x window repacked the same way      -> B pairs contiguous
//  Conv: 8 waves = 8 tiles (4 f-tiles x 2 x-tiles), rolled K loop,
//        body = ds_load_b64 (A) + ds_load_b64 (B) + v_wmma_f32_16x16x4_f32,
//        two accumulator chains to break the D->C dependency.
//  Histogram: linear LDS streams over the same wt/xb buffers (no global reads).
// ---------------------------------------------------------------------------
__global__ __launch_bounds__(256)
void fused_conv_hist(const float* __restrict__ x,
                     const float* __restrict__ w,
                     float* __restrict__ out,
                     unsigned int* __restrict__ ghist)
{
    __shared__ __align__(16) float wt_s[F_OUT * KDIM];   // 147456 B
    __shared__ __align__(16) float xb_s[W_IMG * KDIM];   //  73728 B
    __shared__ unsigned int s_hist[NBINS];

    const int tid = threadIdx.x;
    const int row = blockIdx.x;           // 0..255 = n*32 + y
    const int y = row & 31;
    const int n = row >> 5;

    if (tid < NBINS) s_hist[tid] = 0u;

    // ---- Phase 1a: stage w, repacked K-major (all cells valid -> all async)
    {
        const unsigned wt_base = (unsigned)(uintptr_t)&wt_s[0];
        for (int idx = tid; idx < F_OUT * KDIM; idx += 256) {
            const int f   = idx / KDIM;          // coalesced global read order
            const int rem = idx - f * KDIM;
            const int c   = rem / 9;
            const int tap = rem - c * 9;
            const unsigned dst = (unsigned)(f * KDIM + tap * 64 + c);
            async_f32_to_lds(wt_base + dst * 4u, w + idx);
        }
    }

    // ---- Phase 1b: stage padded x window, repacked K-major -----------------
    // xb[xcol][tap*64 + c] = xpad[n, c, y + i - 1, xcol + j - 1]
    // valid cells: async global->LDS; out-of-image cells: ds_store 0 (disjoint)
    {
        const unsigned xb_base = (unsigned)(uintptr_t)&xb_s[0];
        for (int idx = tid; idx < W_IMG * KDIM; idx += 256) {
            const int xcol = idx / KDIM;
            const int k    = idx - xcol * KDIM;
            const int tap  = k >> 6;
            const int c    = k & 63;
            const int i = tap / 3;
            const int j = tap - i * 3;
            const int yy = y + i - 1;
            const int xx = xcol + j - 1;
            if (yy >= 0 && yy < H_IMG && xx >= 0 && xx < W_IMG) {
                async_f32_to_lds(xb_base + (unsigned)idx * 4u,
                    x + ((n * C_IN + c) * H_IMG + yy) * W_IMG + xx);
            } else {
                xb_s[idx] = 0.f;
            }
        }
    }
    wait_asynccnt0();
    __syncthreads();

    // ---- Phase 2: conv via V_WMMA_F32_16X16X4_F32 --------------------------
    {
        const int wv   = tid >> 5;
        const int lane = tid & 31;
        const int ft = wv & 3;            // f-tile 0..3
        const int xt = wv >> 2;           // 16-pixel half row 0..1

        const int m     = lane & 15;      // A row (f) / B col (pixel) in tile
        const int khalf = lane >> 4;      // lanes 16-31 carry K slots {2,3}
        const int f     = ft * 16 + m;
        const int xcol  = xt * 16 + m;

        const int aoff = f * KDIM + khalf * 2;     // float index, 8B aligned
        const int boff = xcol * KDIM + khalf * 2;

        v8f acc0 = {0.f, 0.f, 0.f, 0.f, 0.f, 0.f, 0.f, 0.f};
        v8f acc1 = {0.f, 0.f, 0.f, 0.f, 0.f, 0.f, 0.f, 0.f};

#pragma unroll 4
        for (int kk = 0; kk < KDIM / 4; kk += 2) {
            v2f a0 = *(const v2f*)&wt_s[aoff + kk * 4];
            v2f b0 = *(const v2f*)&xb_s[boff + kk * 4];
            acc0 = __builtin_amdgcn_wmma_f32_16x16x4_f32(
                false, a0, false, b0, (short)0, acc0, false, false);
            v2f a1 = *(const v2f*)&wt_s[aoff + kk * 4 + 4];
            v2f b1 = *(const v2f*)&xb_s[boff + kk * 4 + 4];
            acc1 = __builtin_amdgcn_wmma_f32_16x16x4_f32(
                false, a1, false, b1, (short)0, acc1, false, false);
        }
        acc0 = acc0 + acc1;

        // D layout: VGPR r -> M = r (+8 for lanes 16..31), N = lane&15
        const int fbase = ft * 16 + khalf * 8;
#pragma unroll
        for (int r = 0; r < 8; ++r) {
            const int fo = fbase + r;
            out[((n * F_OUT + fo) * H_IMG + y) * W_IMG + xt * 16 + m] = acc0[r];
        }
    }

    // ---- Phase 3: histogram over the same LDS buffers ----------------------
    // k = tap*64 + g*8 + cc: each run of 8 consecutive k is one (tap, group)
    // => both operand reads are linear LDS streams.
    // Packed counters: bins 0..4 in 12-bit fields (max 576 <= 4095),
    // bins 5..8 in 16-bit fields.
    {
        const int xcol  = tid & 31;
        const int fbase = tid >> 5;
        const int bx = xcol * KDIM;

        unsigned long long lo = 0ull;
        unsigned long long hi = 0ull;

        for (int it = 0; it < 8; ++it) {
            const int bw = (fbase + 8 * it) * KDIM;
#pragma unroll 8
            for (int kq = 0; kq < KDIM / 8; ++kq) {
                int cnt = 0;
#pragma unroll
                for (int u = 0; u < 8; ++u) {
                    // |w| * |x| -> single v_mul with abs src modifiers
                    const float p = fabsf(wt_s[bw + kq * 8 + u]) *
                                    fabsf(xb_s[bx + kq * 8 + u]);
                    cnt += (p > EPS_F) ? 1 : 0;
                }
                if (cnt < 5) lo += 1ull << (12 * cnt);
                else         hi += 1ull << (16 * (cnt - 5));
            }
        }

#pragma unroll
        for (int k = 0; k < 5; ++k)
            atomicAdd(&s_hist[k], (unsigned int)((lo >> (12 * k)) & 0xFFFull));
#pragma unroll
        for (int k = 0; k < 4; ++k)
            atomicAdd(&s_hist[5 + k], (unsigned int)((hi >> (16 * k)) & 0xFFFFull));
        __syncthreads();

        if (tid < NBINS) atomicAdd(&ghist[tid], s_hist[tid]);
    }
}

// ---------------------------------------------------------------------------
// Graph-capture-safe init / finalize (harness does not re-zero d_ws)
// ---------------------------------------------------------------------------
__global__ void zero_hist(unsigned int* h)
{
    if (threadIdx.x < NBINS) h[threadIdx.x] = 0u;
}

__global__ void finalize_hist(const unsigned int* __restrict__ h,
                              float* __restrict__ out_hist)
{
    if (threadIdx.x < NBINS) out_hist[threadIdx.x] = (float)h[threadIdx.x];
}

// ---------------------------------------------------------------------------
extern "C" void kernel_launch(void* const* d_in, const int* in_sizes, int n_in,
                              void* d_out, int out_size, void* d_ws, size_t ws_size,
                              hipStream_t stream)
{
    (void)in_sizes; (void)n_in; (void)out_size; (void)ws_size;
    const float* x = (const float*)d_in[0];   // [8,64,32,32]
    const float* w = (const float*)d_in[1];   // [64,64,3,3]
    // d_in[2] = bias, unused by the reference math
    float* out = (float*)d_out;               // conv out + 9 hist floats
    unsigned int* hist = (unsigned int*)d_ws;

    zero_hist<<<1, 32, 0, stream>>>(hist);
    fused_conv_hist<<<N_BATCH * H_IMG, 256, 0, stream>>>(x, w, out, hist);
    finalize_hist<<<1, 32, 0, stream>>>(hist, out + OUT_ELEMS);
}